// SimpleGNNLayer_78512002171437
// MI455X (gfx1250) — compile-verified
//
#include <hip/hip_runtime.h>

#define NN 50000
#define EE 800000
#define HH 8
#define CC 32
#define HSZ 256
#define NEG_SLOPE 0.2f
#define EPSV 1e-16f

typedef __attribute__((ext_vector_type(2))) float v2f;
typedef __attribute__((ext_vector_type(8))) float v8f;

// -------- workspace layout (bytes) --------
#define OFF_XW   ((size_t)0)                              // [N,256] f32  = 51.2 MB
#define OFF_ASRC (OFF_XW   + (size_t)NN * HSZ * 4)        // [N,8]  f32
#define OFF_ADST (OFF_ASRC + (size_t)NN * HH * 4)         // [N,8]  f32
#define OFF_MENC (OFF_ADST + (size_t)NN * HH * 4)         // [N,8]  u32 (encoded max)
#define OFF_DEN  (OFF_MENC + (size_t)NN * HH * 4)         // [N,8]  f32
#define OFF_EBUF (OFF_DEN  + (size_t)NN * HH * 4)         // [E,8]  f32 = 25.6 MB

// order-preserving float <-> uint encoding for atomic max
__device__ __forceinline__ unsigned enc_f32(float f) {
  unsigned u = __float_as_uint(f);
  return (u & 0x80000000u) ? ~u : (u | 0x80000000u);
}
__device__ __forceinline__ float dec_f32(unsigned u) {
  unsigned b = (u & 0x80000000u) ? (u & 0x7fffffffu) : ~u;
  return __uint_as_float(b);
}

__device__ __forceinline__ void atomAddF32(float* p, float v) {
  unsafeAtomicAdd(p, v);   // native GLOBAL_ATOMIC_ADD_F32 (no CAS loop)
}

// -------- 0) zero-init out / max / denom --------
__global__ void __launch_bounds__(256) gat_init(float* __restrict__ out,
                                                unsigned* __restrict__ menc,
                                                float* __restrict__ den) {
  int i = blockIdx.x * 256 + threadIdx.x;          // grid covers N*256 exactly
  out[i] = 0.0f;
  if (i < NN * HH) { menc[i] = 0u; den[i] = 0.0f; }
}

// -------- 1) xw = x @ W with V_WMMA_F32_16X16X4_F32 --------
// block = 256 thr (8 waves) -> one 16-row stripe; wave w owns col-tiles {w, w+8}
__global__ void __launch_bounds__(256) gat_gemm(const float* __restrict__ x,
                                                const float* __restrict__ W,
                                                float* __restrict__ xw) {
  const int rowBase = blockIdx.x * 16;
  const int wave = threadIdx.x >> 5;
  const int lane = threadIdx.x & 31;
  const int half = lane >> 4;        // 0: K=k,k+1  1: K=k+2,k+3 (fp32 A layout)
  const int lrow = lane & 15;
  const int col0 = wave * 16 + lrow;
  const int col1 = (wave + 8) * 16 + lrow;

  const float* arow = x + (size_t)(rowBase + lrow) * HSZ + half * 2;
  v8f c0 = {};
  v8f c1 = {};
#pragma unroll 8
  for (int k = 0; k < HSZ; k += 4) {
    v2f a = *(const v2f*)(arow + k);                       // A[M=lrow, kk..kk+1]
    const float* wk = W + (size_t)(k + half * 2) * HSZ;    // B rows kk, kk+1
    v2f b0, b1;
    b0.x = wk[col0];       b0.y = wk[HSZ + col0];
    b1.x = wk[col1];       b1.y = wk[HSZ + col1];
    c0 = __builtin_amdgcn_wmma_f32_16x16x4_f32(false, a, false, b0,
                                               (short)0, c0, false, false);
    c1 = __builtin_amdgcn_wmma_f32_16x16x4_f32(false, a, false, b1,
                                               (short)0, c1, false, false);
  }
  // D layout: VGPR r -> M = r + half*8, N = lrow
  const int rowHalf = rowBase + half * 8;
#pragma unroll
  for (int r = 0; r < 8; ++r) {
    xw[(size_t)(rowHalf + r) * HSZ + col0] = c0[r];
    xw[(size_t)(rowHalf + r) * HSZ + col1] = c1[r];
  }
}

// -------- 2) a_src[n,h] = <xw[n,h,:], att_src[h,:]> (and dst) --------
// one wave per node; lane l covers channels [8l, 8l+8) (stays inside one head)
__global__ void __launch_bounds__(256) gat_att(const float* __restrict__ xw,
                                               const float* __restrict__ att_src,
                                               const float* __restrict__ att_dst,
                                               float* __restrict__ a_src,
                                               float* __restrict__ a_dst) {
  const int node = blockIdx.x * 8 + (threadIdx.x >> 5);
  const int lane = threadIdx.x & 31;
  const float4* xr = (const float4*)(xw + (size_t)node * HSZ + lane * 8);
  const float4* as = (const float4*)(att_src + lane * 8);
  const float4* ad = (const float4*)(att_dst + lane * 8);
  float4 v0 = xr[0], v1 = xr[1];
  float4 s0 = as[0], s1 = as[1];
  float4 d0 = ad[0], d1 = ad[1];
  float ss = v0.x * s0.x + v0.y * s0.y + v0.z * s0.z + v0.w * s0.w +
             v1.x * s1.x + v1.y * s1.y + v1.z * s1.z + v1.w * s1.w;
  float sd = v0.x * d0.x + v0.y * d0.y + v0.z * d0.z + v0.w * d0.w +
             v1.x * d1.x + v1.y * d1.y + v1.z * d1.z + v1.w * d1.w;
  // reduce groups of 4 lanes -> one head each
  ss += __shfl_xor(ss, 1, 32); ss += __shfl_xor(ss, 2, 32);
  sd += __shfl_xor(sd, 1, 32); sd += __shfl_xor(sd, 2, 32);
  if ((lane & 3) == 0) {
    int h = lane >> 2;
    a_src[node * HH + h] = ss;
    a_dst[node * HH + h] = sd;
  }
}

// -------- 3) e = leaky_relu(a_src[src]+a_dst[dst]); segment max over dst -----
__global__ void __launch_bounds__(256) gat_edge_max(const int* __restrict__ ei,
                                                    const float* __restrict__ a_src,
                                                    const float* __restrict__ a_dst,
                                                    float* __restrict__ ebuf,
                                                    unsigned* __restrict__ menc) {
  int tid = blockIdx.x * 256 + threadIdx.x;  // E*H = 6.4M
  int e = tid >> 3, h = tid & 7;
  int src = ei[e], dst = ei[EE + e];
  float v = a_src[src * HH + h] + a_dst[dst * HH + h];
  v = v > 0.0f ? v : NEG_SLOPE * v;
  ebuf[tid] = v;
  atomicMax(&menc[dst * HH + h], enc_f32(v));
}

// -------- 4) ex = exp(e - m[dst]); denom = segment_sum(ex) --------
__global__ void __launch_bounds__(256) gat_edge_exp(const int* __restrict__ ei,
                                                    float* __restrict__ ebuf,
                                                    const unsigned* __restrict__ menc,
                                                    float* __restrict__ den) {
  int tid = blockIdx.x * 256 + threadIdx.x;
  int e = tid >> 3, h = tid & 7;
  int dst = ei[EE + e];
  float m = dec_f32(menc[dst * HH + h]);      // always non-empty here
  float ex = __expf(ebuf[tid] - m);
  ebuf[tid] = ex;
  atomAddF32(&den[dst * HH + h], ex);
}

// -------- 5) out[dst] += xw[src] * alpha   (one block = one edge, 256 ch) ----
__global__ void __launch_bounds__(256) gat_edge_msg(const int* __restrict__ ei,
                                                    const float* __restrict__ ebuf,
                                                    const float* __restrict__ den,
                                                    const float* __restrict__ xw,
                                                    float* __restrict__ out) {
  int e = blockIdx.x;
  int c = threadIdx.x;
  int h = c >> 5;
  int src = ei[e], dst = ei[EE + e];
  float alpha = ebuf[e * HH + h] / (den[dst * HH + h] + EPSV);
  float v = xw[(size_t)src * HSZ + c] * alpha;
  atomAddF32(&out[(size_t)dst * HSZ + c], v);
}

// -------- 6) out = relu(out + bias) --------
__global__ void __launch_bounds__(256) gat_final(float* __restrict__ out,
                                                 const float* __restrict__ bias) {
  int i = blockIdx.x * 256 + threadIdx.x;     // grid covers N*256 exactly
  float v = out[i] + bias[i & (HSZ - 1)];
  out[i] = v > 0.0f ? v : 0.0f;
}

extern "C" void kernel_launch(void* const* d_in, const int* in_sizes, int n_in,
                              void* d_out, int out_size, void* d_ws, size_t ws_size,
                              hipStream_t stream) {
  const float* x       = (const float*)d_in[0];
  const int*   ei      = (const int*)d_in[1];   // [2,E] int32 (JAX x64-off)
  const float* W       = (const float*)d_in[2];
  const float* att_src = (const float*)d_in[3];
  const float* att_dst = (const float*)d_in[4];
  const float* bias    = (const float*)d_in[5];
  float* out = (float*)d_out;

  char* ws = (char*)d_ws;
  float*    xw    = (float*)(ws + OFF_XW);
  float*    a_src = (float*)(ws + OFF_ASRC);
  float*    a_dst = (float*)(ws + OFF_ADST);
  unsigned* menc  = (unsigned*)(ws + OFF_MENC);
  float*    den   = (float*)(ws + OFF_DEN);
  float*    ebuf  = (float*)(ws + OFF_EBUF);

  gat_init<<<NN * HSZ / 256, 256, 0, stream>>>(out, menc, den);          // 50000 blk
  gat_gemm<<<NN / 16, 256, 0, stream>>>(x, W, xw);                       // 3125 blk
  gat_att<<<NN / 8, 256, 0, stream>>>(xw, att_src, att_dst, a_src, a_dst);
  gat_edge_max<<<EE * HH / 256, 256, 0, stream>>>(ei, a_src, a_dst, ebuf, menc);
  gat_edge_exp<<<EE * HH / 256, 256, 0, stream>>>(ei, ebuf, menc, den);
  gat_edge_msg<<<EE, 256, 0, stream>>>(ei, ebuf, den, xw, out);
  gat_final<<<NN * HSZ / 256, 256, 0, stream>>>(out, bias);
}